// Shape2DPainterNet_22333829940016
// MI455X (gfx1250) — compile-verified
//
#include <hip/hip_runtime.h>
#include <math.h>

typedef float v2f __attribute__((ext_vector_type(2)));
typedef float v8f __attribute__((ext_vector_type(8)));

#define HDIM 64
#define TSTEPS 128
#define NCANVAS 25
#define ROWS 64      // batch rows per workgroup
#define PITCH 68     // LDS row pitch for X/H tiles (bank-conflict free, 16B aligned)

__device__ __forceinline__ float sigmoidf_(float x) { return 1.0f / (1.0f + __expf(-x)); }

__global__ __launch_bounds__(256)
void painter_lstm_kernel(const int* __restrict__ inst,
                         const float* __restrict__ prev_canvas,
                         const float* __restrict__ embed,
                         const float* __restrict__ W_ih,
                         const float* __restrict__ W_hh,
                         const float* __restrict__ b_ih,
                         const float* __restrict__ b_hh,
                         const float* __restrict__ Wc, const float* __restrict__ bc,
                         const float* __restrict__ Ws, const float* __restrict__ bs,
                         const float* __restrict__ Wo, const float* __restrict__ bo,
                         const float* __restrict__ Wr1, const float* __restrict__ br1,
                         const float* __restrict__ Wr2, const float* __restrict__ br2,
                         float* __restrict__ out, int B)
{
    // K-pair interleaved weights: W[(k>>1)*512 + n*2 + (k&1)] = Worig[n*64 + k]
    __shared__ float WI[64 * 256];     // 64 KB
    __shared__ float WH[64 * 256];     // 64 KB
    __shared__ float XL[ROWS * PITCH]; // x tile
    __shared__ float HL[ROWS * PITCH]; // h tile (persistent across steps)
    __shared__ float SC[ROWS * 32];    // canvas scores
    __shared__ float R1[32 * 69];      // Wr1
    __shared__ float RB1[32];
    __shared__ float WR2[32];
    __shared__ float TAG[ROWS];

    const int tid  = threadIdx.x;
    const int lane = tid & 31;
    const int wv   = tid >> 5;        // wave 0..7
    const int l16  = lane & 15;
    const int hi   = lane >> 4;       // lane half
    const int ko   = hi << 1;         // K-select: lanes 0-15 -> K+{0,1}, 16-31 -> K+{2,3}
    const int hc   = wv & 3;          // hidden-col tile (16 cols) owned by this wave
    const int rg   = (wv >> 2) << 1;  // row-tile base: 0 or 2
    const int rbase = blockIdx.x * ROWS;

    // ---- stage weights into LDS (once) ----
    for (int i = tid; i < 64 * 256; i += 256) {
        int n = i >> 6, k = i & 63;
        int d = ((k >> 1) << 9) + (n << 1) + (k & 1);
        WI[d] = W_ih[i];
        WH[d] = W_hh[i];
    }
    for (int i = tid; i < ROWS * PITCH; i += 256) HL[i] = 0.0f;

    // per-lane fused biases for the 4 gates at col = q*64 + hc*16 + l16
    const int cb = hc * 16 + l16;
    const float bI = b_ih[cb]       + b_hh[cb];
    const float bF = b_ih[cb + 64]  + b_hh[cb + 64];
    const float bG = b_ih[cb + 128] + b_hh[cb + 128];
    const float bO = b_ih[cb + 192] + b_hh[cb + 192];

    // persistent cell state in registers (2 row-tiles x 8 C-layout elems)
    float cst[2][8];
    #pragma unroll
    for (int r = 0; r < 2; ++r)
        #pragma unroll
        for (int j = 0; j < 8; ++j) cst[r][j] = 0.0f;

    __syncthreads();

    const int grow = tid >> 2;        // gather: 4 threads per row
    const int gq   = (tid & 3) * 16;  // 16-float segment of the embedding row

    for (int t = 0; t < TSTEPS; ++t) {
        // ---- gather x_t = embed[inst[row, t]] into XL ----
        {
            int idx = inst[(rbase + grow) * TSTEPS + t];
            const float4* src = (const float4*)(embed + idx * HDIM + gq);
            float4* dst = (float4*)(XL + grow * PITCH + gq);
            dst[0] = src[0]; dst[1] = src[1]; dst[2] = src[2]; dst[3] = src[3];
        }
        __syncthreads();

        // ---- gates = x W_ih^T + h W_hh^T + b  via V_WMMA_F32_16X16X4_F32 ----
        v8f accs[2][4];
        #pragma unroll
        for (int rr = 0; rr < 2; ++rr) {
            const int r = rg + rr;
            const int m = r * 16 + l16;
            const float* xrow = XL + m * PITCH;
            const float* hrow = HL + m * PITCH;

            v8f aI = {bI, bI, bI, bI, bI, bI, bI, bI};
            v8f aF = {bF, bF, bF, bF, bF, bF, bF, bF};
            v8f aG = {bG, bG, bG, bG, bG, bG, bG, bG};
            v8f aO = {bO, bO, bO, bO, bO, bO, bO, bO};

            #pragma unroll
            for (int kb = 0; kb < HDIM; kb += 4) {
                v2f a = *(const v2f*)(xrow + kb + ko);
                const float* wk = WI + (((kb + ko) >> 1) << 9);
                v2f b0 = *(const v2f*)(wk + ((0   + cb) << 1));
                v2f b1 = *(const v2f*)(wk + ((64  + cb) << 1));
                v2f b2 = *(const v2f*)(wk + ((128 + cb) << 1));
                v2f b3 = *(const v2f*)(wk + ((192 + cb) << 1));
                aI = __builtin_amdgcn_wmma_f32_16x16x4_f32(false, a, false, b0, (short)0, aI, false, false);
                aF = __builtin_amdgcn_wmma_f32_16x16x4_f32(false, a, false, b1, (short)0, aF, false, false);
                aG = __builtin_amdgcn_wmma_f32_16x16x4_f32(false, a, false, b2, (short)0, aG, false, false);
                aO = __builtin_amdgcn_wmma_f32_16x16x4_f32(false, a, false, b3, (short)0, aO, false, false);
            }
            #pragma unroll
            for (int kb = 0; kb < HDIM; kb += 4) {
                v2f a = *(const v2f*)(hrow + kb + ko);
                const float* wk = WH + (((kb + ko) >> 1) << 9);
                v2f b0 = *(const v2f*)(wk + ((0   + cb) << 1));
                v2f b1 = *(const v2f*)(wk + ((64  + cb) << 1));
                v2f b2 = *(const v2f*)(wk + ((128 + cb) << 1));
                v2f b3 = *(const v2f*)(wk + ((192 + cb) << 1));
                aI = __builtin_amdgcn_wmma_f32_16x16x4_f32(false, a, false, b0, (short)0, aI, false, false);
                aF = __builtin_amdgcn_wmma_f32_16x16x4_f32(false, a, false, b1, (short)0, aF, false, false);
                aG = __builtin_amdgcn_wmma_f32_16x16x4_f32(false, a, false, b2, (short)0, aG, false, false);
                aO = __builtin_amdgcn_wmma_f32_16x16x4_f32(false, a, false, b3, (short)0, aO, false, false);
            }
            accs[rr][0] = aI; accs[rr][1] = aF; accs[rr][2] = aG; accs[rr][3] = aO;
        }
        __syncthreads();   // all GEMM reads of XL/HL complete

        // ---- LSTM elementwise on WMMA C-layout, update c, write h ----
        #pragma unroll
        for (int rr = 0; rr < 2; ++rr) {
            const int r = rg + rr;
            #pragma unroll
            for (int j = 0; j < 8; ++j) {
                float iv = sigmoidf_(accs[rr][0][j]);
                float fv = sigmoidf_(accs[rr][1][j]);
                float gv = tanhf(accs[rr][2][j]);
                float ov = sigmoidf_(accs[rr][3][j]);
                float c  = fv * cst[rr][j] + iv * gv;
                cst[rr][j] = c;
                HL[(r * 16 + j + hi * 8) * PITCH + hc * 16 + l16] = ov * tanhf(c);
            }
        }
        // next iteration's post-gather barrier orders HL writes vs. reads
    }
    __syncthreads();

    // ---- stage scorer weights + action tag ----
    for (int i = tid; i < 32 * 69; i += 256) R1[i] = Wr1[i];
    if (tid < 32) { RB1[tid] = br1[tid]; WR2[tid] = Wr2[tid]; }
    if (tid < ROWS) {
        int last = -1;
        const int* ip = inst + (rbase + tid) * TSTEPS;
        for (int t = 0; t < TSTEPS; ++t) { int v = ip[t]; if (v >= 1) last = v; }
        TAG[tid] = (last == 9) ? 1.0f : 0.0f;
    }
    __syncthreads();

    // ---- canvas scores: one (row, canvas) pair per thread ----
    const float br2v = br2[0];
    for (int p = tid; p < ROWS * NCANVAS; p += 256) {
        int row = p / NCANVAS;
        int n   = p - row * NCANVAS;
        const float* hrow = HL + row * PITCH;
        const float* pcv  = prev_canvas + (rbase + row) * NCANVAS * 4 + n * 4;
        float tg = TAG[row];
        float sc = br2v;
        for (int j = 0; j < 32; ++j) {
            const float* wj = R1 + j * 69;
            float acc = RB1[j];
            #pragma unroll 8
            for (int k = 0; k < HDIM; ++k) acc += wj[k] * hrow[k];
            acc += wj[64] * tg + wj[65] * pcv[0] + wj[66] * pcv[1] + wj[67] * pcv[2] + wj[68] * pcv[3];
            sc += WR2[j] * fmaxf(acc, 0.0f);
        }
        SC[row * 32 + n] = sc;
    }
    __syncthreads();

    // ---- per-row: softmax over canvas, heads, outputs ----
    if (tid < ROWS) {
        const int row = tid;
        const int g = rbase + row;
        const float* hrow = HL + row * PITCH;

        float mx = -INFINITY;
        for (int n = 0; n < NCANVAS; ++n) mx = fmaxf(mx, SC[row * 32 + n]);
        float w[NCANVAS]; float ssum = 0.0f;
        for (int n = 0; n < NCANVAS; ++n) { float e = __expf(SC[row * 32 + n] - mx); w[n] = e; ssum += e; }
        float inv = 1.0f / ssum;
        const float* pc = prev_canvas + g * NCANVAS * 4;
        float r2 = 0.0f, r3 = 0.0f;
        for (int n = 0; n < NCANVAS; ++n) { float wn = w[n] * inv; r2 += wn * pc[n * 4 + 2]; r3 += wn * pc[n * 4 + 3]; }

        float o0 = bo[0], o1 = bo[1];
        float c0 = bc[0], c1 = bc[1], c2 = bc[2];
        float s0 = bs[0], s1 = bs[1], s2 = bs[2];
        for (int k = 0; k < HDIM; ++k) {
            float hk = hrow[k];
            o0 += Wo[k] * hk;        o1 += Wo[64 + k] * hk;
            c0 += Wc[k] * hk;        c1 += Wc[64 + k] * hk;  c2 += Wc[128 + k] * hk;
            s0 += Ws[k] * hk;        s1 += Ws[64 + k] * hk;  s2 += Ws[128 + k] * hk;
        }
        o0 = fminf(fmaxf(o0, -1.0f), 1.0f);
        o1 = fminf(fmaxf(o1, -1.0f), 1.0f);

        float cm = fmaxf(c0, fmaxf(c1, c2));
        float cl = __logf(__expf(c0 - cm) + __expf(c1 - cm) + __expf(c2 - cm)) + cm;
        float sm = fmaxf(s0, fmaxf(s1, s2));
        float sl = __logf(__expf(s0 - sm) + __expf(s1 - sm) + __expf(s2 - sm)) + sm;

        out[g * 3 + 0] = c0 - cl; out[g * 3 + 1] = c1 - cl; out[g * 3 + 2] = c2 - cl;
        float* shp = out + (size_t)B * 3;
        shp[g * 3 + 0] = s0 - sl; shp[g * 3 + 1] = s1 - sl; shp[g * 3 + 2] = s2 - sl;
        out[(size_t)B * 6 + g] = r2 + o0;
        out[(size_t)B * 7 + g] = r3 + o1;
    }
}

extern "C" void kernel_launch(void* const* d_in, const int* in_sizes, int n_in,
                              void* d_out, int out_size, void* d_ws, size_t ws_size,
                              hipStream_t stream) {
    const int*   inst  = (const int*)  d_in[0];
    const float* pcv   = (const float*)d_in[1];
    // d_in[2] ref_obj, d_in[3] target_obj: unused by the reference outputs
    const float* embed = (const float*)d_in[4];
    const float* W_ih  = (const float*)d_in[5];
    const float* W_hh  = (const float*)d_in[6];
    const float* b_ih  = (const float*)d_in[7];
    const float* b_hh  = (const float*)d_in[8];
    const float* Wc    = (const float*)d_in[9];
    const float* bc    = (const float*)d_in[10];
    const float* Ws    = (const float*)d_in[11];
    const float* bs    = (const float*)d_in[12];
    const float* Wo    = (const float*)d_in[13];
    const float* bo    = (const float*)d_in[14];
    const float* Wr1   = (const float*)d_in[15];
    const float* br1   = (const float*)d_in[16];
    const float* Wr2   = (const float*)d_in[17];
    const float* br2   = (const float*)d_in[18];

    const int B = in_sizes[0] / TSTEPS;      // 4096
    const int blocks = B / ROWS;             // 64 persistent workgroups

    painter_lstm_kernel<<<blocks, 256, 0, stream>>>(
        inst, pcv, embed, W_ih, W_hh, b_ih, b_hh,
        Wc, bc, Ws, bs, Wo, bo, Wr1, br1, Wr2, br2,
        (float*)d_out, B);
}